// Model_37211596653272
// MI455X (gfx1250) — compile-verified
//
#include <hip/hip_runtime.h>
#include <cstdint>
#include <cstddef>

// MI455X (gfx1250) implementation of the spiking CNN reference.
//
// - One kernel per (timestep, layer): the LIF recurrence forces serialization
//   over t and over layers; kernel boundaries provide the sync.
// - 1x1 convs and FC1 run on the matrix pipe via V_WMMA_F32_16X16X4_F32
//   (fp32-exact w.r.t. the reference; spikes are exact {0,1}).
// - Membranes/spikes live in workspace in NHWC so the WMMA C-fragment epilogue
//   (lanes 0..15 -> consecutive N) does coalesced 64B mem read/modify/write.
// - Spikes stored as u8; 2x2 max-pool of {0,1} fused into the consumer's
//   A-fragment load as a byte-wise OR (each spike byte read exactly once).
// - M/K padding is done with zero-padded buffers / value selects, never with
//   per-lane predicated loads, so inner loops are branch-free load+wmma.

typedef __attribute__((ext_vector_type(2))) float v2f;
typedef __attribute__((ext_vector_type(8))) float v8f;

#define LIF_BETA 0.95f
#define LIF_THR  1.0f
#define BATCH    8

// ---------------------------------------------------------------------------
// Conv1x1 (channel contraction) + LIF update, WMMA f32 16x16x4.
//   M = 16 pixels (flattened B*H*W), N = 16 out-channels, K = in-channels.
// A fragment (16x4 f32): lanes 0-15 hold M=lane, K = {0,1}; lanes 16-31 hold
// M=lane-16, K = {2,3}. B fragment mapped symmetrically (N=lane&15).
// C/D fragment: VGPR r, lanes 0-15 -> (M=r, N=lane), lanes 16-31 -> (M=r+8).
// ---------------------------------------------------------------------------
template <int CIN, int COUT, int H, int W, bool FIRST>
__global__ __launch_bounds__(256) void lif_conv_layer(
    const float* __restrict__ xin,     // FIRST: x_t  [B,2,H,W] fp32 (planar)
    const uint8_t* __restrict__ sin,   // !FIRST: spikes [B,2H,2W,CIN] u8 NHWC
    const float* __restrict__ wgt,     // [COUT, CIN]
    const float* __restrict__ bias,    // [COUT]
    float* __restrict__ mem,           // [B*H*W, COUT] NHWC, persistent state
    uint8_t* __restrict__ spk)         // [B*H*W, COUT] NHWC, prev/next spikes
{
    constexpr int PIX    = BATCH * H * W;
    constexpr int NTILES = COUT / 16;
    constexpr int CHUNKS = (CIN + 3) / 4;

    const int wave = blockIdx.x * (blockDim.x >> 5) + (threadIdx.x >> 5);
    if (wave >= PIX / 16) return;            // wave-uniform: EXEC stays full
    const int lane = threadIdx.x & 31;
    const int mrow = lane & 15;
    const int half = lane >> 4;              // K offset base = 2*half
    const bool k_lo = (half == 0);
    const int p    = wave * 16 + mrow;       // this lane's A-row pixel

    // ---- load A fragments once; reused across all N tiles ----
    v2f afrag[CHUNKS];
    if constexpr (FIRST) {
        constexpr int HW = H * W;
        const int b = p / HW;
        const int r = p - b * HW;
        const float* xp = xin + (size_t)b * 2 * HW + r;   // [B,2,H,W] planar
        // unconditional loads (addresses valid for all lanes), then select:
        const float c0 = xp[0];
        const float c1 = xp[HW];
        v2f a;
        a.x = k_lo ? c0 : 0.0f;              // K=0 | K=2 pad
        a.y = k_lo ? c1 : 0.0f;              // K=1 | K=3 pad
        afrag[0] = a;
    } else {
        constexpr int HW = H * W;
        const int b  = p / HW;
        const int r  = p - b * HW;
        const int h  = r / W;
        const int wc = r - h * W;
        constexpr int IW = 2 * W;
        const uint8_t* sp =
            sin + ((((size_t)b * (2 * H) + 2 * h) * IW) + 2 * wc) * CIN;
        #pragma unroll
        for (int ck = 0; ck < CHUNKS; ++ck) {
            const int k = ck * 4 + half * 2;          // even -> 2B aligned
            unsigned v0 = *(const unsigned short*)(sp + k);
            unsigned v1 = *(const unsigned short*)(sp + CIN + k);
            unsigned v2 = *(const unsigned short*)(sp + (size_t)IW * CIN + k);
            unsigned v3 = *(const unsigned short*)(sp + (size_t)IW * CIN + CIN + k);
            unsigned v  = v0 | v1 | v2 | v3;          // max of {0,1} == OR
            v2f a;
            a.x = (float)(v & 0xFFu);
            a.y = (float)((v >> 8) & 0xFFu);
            afrag[ck] = a;
        }
    }

    const int ncol = lane & 15;
    #pragma unroll 1
    for (int nt = 0; nt < NTILES; ++nt) {
        const int n = nt * 16 + ncol;
        const float* wp = wgt + (size_t)n * CIN;
        v8f acc = {};
        #pragma unroll
        for (int ck = 0; ck < CHUNKS; ++ck) {
            v2f bfr;
            if constexpr (FIRST) {
                const float w0 = wp[0];
                const float w1 = wp[1];
                bfr.x = k_lo ? w0 : 0.0f;
                bfr.y = k_lo ? w1 : 0.0f;
            } else {
                const int k = ck * 4 + half * 2;      // even -> 8B aligned
                bfr = *(const v2f*)(wp + k);
            }
            acc = __builtin_amdgcn_wmma_f32_16x16x4_f32(
                false, afrag[ck], false, bfr, (short)0, acc, false, false);
        }
        // ---- LIF epilogue: coalesced NHWC (lanes 0..15 -> consecutive n) ----
        const float bv = bias[n];
        #pragma unroll
        for (int r8 = 0; r8 < 8; ++r8) {
            const int em = r8 + half * 8;
            const size_t idx = (size_t)(wave * 16 + em) * COUT + n;
            const float cur  = acc[r8] + bv;
            const float old  = mem[idx];
            const float prev = (float)spk[idx];                 // spk_{t-1}
            const float nm   = LIF_BETA * old + cur - prev * LIF_THR;
            mem[idx] = nm;
            spk[idx] = (nm - LIF_THR > 0.0f) ? (uint8_t)1 : (uint8_t)0;
        }
    }
}

// ---------------------------------------------------------------------------
// Pool layer-5 spikes [B,16,20,128] NHWC -> accumulate into s[B, c*80+h*10+w]
// (reference's channel-major flatten order for the FC head).  s is allocated
// as [16, 10240] with rows 8..15 kept zero (M padding for the FC1 WMMA tile).
// ---------------------------------------------------------------------------
__global__ __launch_bounds__(256) void pool_accum_head(
    const uint8_t* __restrict__ spk5, float* __restrict__ s)
{
    const int tid = blockIdx.x * blockDim.x + threadIdx.x;
    if (tid >= BATCH * 128 * 8 * 10) return;
    const int c = tid & 127;       // fast dim -> coalesced u8 reads
    int t = tid >> 7;
    const int w = t % 10; t /= 10;
    const int h = t % 8;  t /= 8;
    const int b = t;
    const uint8_t* p0 = spk5 + (((size_t)(b * 16 + 2 * h) * 20 + 2 * w) * 128) + c;
    const unsigned v = p0[0] | p0[128] | p0[20 * 128] | p0[20 * 128 + 128];
    s[(size_t)b * 10240 + (size_t)c * 80 + h * 10 + w] += (float)v;
}

// ---------------------------------------------------------------------------
// FC1: mem1 = s @ fc1_w^T + b ; spk1 = (mem1 - 1 > 0).  M=16 (rows 8..15 are
// zero padding), N=4096 (256 wave-tiles), K=10240.  fc1_w (168 MB) streamed
// exactly once; inner loop is branch-free b64 loads + wmma.
// ---------------------------------------------------------------------------
__global__ __launch_bounds__(256) void fc1_kernel(
    const float* __restrict__ s,      // [16, 10240], rows 8..15 zero
    const float* __restrict__ w,      // [4096, 10240]
    const float* __restrict__ bias,   // [4096]
    float* __restrict__ spk1)         // [16, 4096] fp32 {0,1}; rows 0..7 used
{
    const int wave = blockIdx.x * (blockDim.x >> 5) + (threadIdx.x >> 5);
    if (wave >= 256) return;
    const int lane = threadIdx.x & 31;
    const int mrow = lane & 15;
    const int half = lane >> 4;
    const int ncol = lane & 15;
    const int n = wave * 16 + ncol;
    const float* sp = s + (size_t)mrow * 10240 + half * 2;
    const float* wp = w + (size_t)n * 10240 + half * 2;
    v8f acc = {};
    #pragma unroll 8
    for (int k0 = 0; k0 < 10240; k0 += 4) {
        const v2f a   = *(const v2f*)(sp + k0);   // 8B-aligned, unconditional
        const v2f bfr = *(const v2f*)(wp + k0);
        acc = __builtin_amdgcn_wmma_f32_16x16x4_f32(
            false, a, false, bfr, (short)0, acc, false, false);
    }
    const float bv = bias[n];
    #pragma unroll
    for (int r = 0; r < 8; ++r) {
        const int em = r + half * 8;              // 0..15; rows 8..15 = pad
        const float m1 = acc[r] + bv;
        spk1[(size_t)em * 4096 + n] = (m1 - 1.0f > 0.0f) ? 1.0f : 0.0f;
    }
}

// ---------------------------------------------------------------------------
// FC2: out[b,o] = spk1[b,:] . fc2_w[o,:] + fc2_b[o]   (32 dot products)
// One wave32 per output, shuffle reduction.
// ---------------------------------------------------------------------------
__global__ __launch_bounds__(1024) void fc2_kernel(
    const float* __restrict__ spk1, const float* __restrict__ w,
    const float* __restrict__ bias, float* __restrict__ out)
{
    const int wave = threadIdx.x >> 5;   // 0..31
    const int lane = threadIdx.x & 31;
    const int b = wave >> 2, o = wave & 3;
    const float* a  = spk1 + (size_t)b * 4096;
    const float* ww = w + (size_t)o * 4096;
    float sum = 0.0f;
    for (int j = lane; j < 4096; j += 32) sum += a[j] * ww[j];
    #pragma unroll
    for (int off = 16; off > 0; off >>= 1) sum += __shfl_xor(sum, off, 32);
    if (lane == 0) out[b * 4 + o] = sum + bias[o];
}

// ---------------------------------------------------------------------------
extern "C" void kernel_launch(void* const* d_in, const int* in_sizes, int n_in,
                              void* d_out, int out_size, void* d_ws, size_t ws_size,
                              hipStream_t stream) {
    const float* x     = (const float*)d_in[0];   // [10,8,2,256,320]
    const float* w1    = (const float*)d_in[1];
    const float* b1    = (const float*)d_in[2];
    const float* w2    = (const float*)d_in[3];
    const float* b2    = (const float*)d_in[4];
    const float* w3    = (const float*)d_in[5];
    const float* b3    = (const float*)d_in[6];
    const float* w4    = (const float*)d_in[7];
    const float* b4    = (const float*)d_in[8];
    const float* w5    = (const float*)d_in[9];
    const float* b5    = (const float*)d_in[10];
    const float* fc1w  = (const float*)d_in[11];
    const float* fc1b  = (const float*)d_in[12];
    const float* fc2w  = (const float*)d_in[13];
    const float* fc2b  = (const float*)d_in[14];
    float* out = (float*)d_out;                   // [8,4]

    // -------- workspace layout (NHWC state buffers) --------
    uint8_t* ws = (uint8_t*)d_ws;
    size_t off = 0;
    auto take = [&](size_t bytes) -> void* {
        void* p = ws + off;
        off = (off + bytes + 255) & ~(size_t)255;
        return p;
    };
    float*   mem1 = (float*)take((size_t)8 * 256 * 320 * 32  * 4);
    float*   mem2 = (float*)take((size_t)8 * 128 * 160 * 64  * 4);
    float*   mem3 = (float*)take((size_t)8 * 64  * 80  * 64  * 4);
    float*   mem4 = (float*)take((size_t)8 * 32  * 40  * 128 * 4);
    float*   mem5 = (float*)take((size_t)8 * 16  * 20  * 128 * 4);
    uint8_t* spk1 = (uint8_t*)take((size_t)8 * 256 * 320 * 32);
    uint8_t* spk2 = (uint8_t*)take((size_t)8 * 128 * 160 * 64);
    uint8_t* spk3 = (uint8_t*)take((size_t)8 * 32  * 40  * 128 / 2 * 4); // 8*64*80*64
    uint8_t* spk4 = (uint8_t*)take((size_t)8 * 32  * 40  * 128);
    uint8_t* spk5 = (uint8_t*)take((size_t)8 * 16  * 20  * 128);
    float*   sacc = (float*)take((size_t)16 * 10240 * 4);   // rows 8..15 zero
    float*   s1f  = (float*)take((size_t)16 * 4096 * 4);    // rows 8..15 pad
    (void)s1f; (void)ws_size; (void)in_sizes; (void)n_in; (void)out_size;

    // zero all persistent state + accumulator + padding once per launch
    hipMemsetAsync(d_ws, 0, off, stream);

    const size_t xstep = (size_t)8 * 2 * 256 * 320;   // elements per timestep
    for (int t = 0; t < 10; ++t) {
        const float* xt = x + (size_t)t * xstep;
        // blocks = (B*H*W/16 waves) / 8 waves-per-block
        lif_conv_layer<2,   32, 256, 320, true ><<<5120, 256, 0, stream>>>(
            xt, nullptr, w1, b1, mem1, spk1);
        lif_conv_layer<32,  64, 128, 160, false><<<1280, 256, 0, stream>>>(
            nullptr, spk1, w2, b2, mem2, spk2);
        lif_conv_layer<64,  64, 64,  80,  false><<<320,  256, 0, stream>>>(
            nullptr, spk2, w3, b3, mem3, spk3);
        lif_conv_layer<64, 128, 32,  40,  false><<<80,   256, 0, stream>>>(
            nullptr, spk3, w4, b4, mem4, spk4);
        lif_conv_layer<128,128, 16,  20,  false><<<20,   256, 0, stream>>>(
            nullptr, spk4, w5, b5, mem5, spk5);
        pool_accum_head<<<320, 256, 0, stream>>>(spk5, sacc);
    }
    fc1_kernel<<<32, 256, 0, stream>>>(sacc, fc1w, fc1b, s1f);
    fc2_kernel<<<1, 1024, 0, stream>>>(s1f, fc2w, fc2b, out);
}